// PoolObj_20590073217622
// MI455X (gfx1250) — compile-verified
//
#include <hip/hip_runtime.h>

#define N_PTS 4096
#define BATCH 4
#define KOUT  2048
#define FEAT  64
#define NBINS 4097
#define R2    (0.05f * 0.05f)
#define RMIN2 (0.02f * 0.02f)

typedef __attribute__((ext_vector_type(2))) float v2f;
typedef __attribute__((ext_vector_type(8))) float v8f;

// ---------------------------------------------------------------------------
// Kernel 1: annulus neighbor counts via V_WMMA_F32_16X16X4_F32 Gram tiles.
// One wave (32 lanes) owns a 16-row tile; loops over all 256 column tiles.
// A layout (16x4 f32): lanes 0-15 -> M=lane, {K0,K1}={x,y}; lanes 16-31 ->
// M=lane-16, {K2,K3}={z,0}.  B mirrors it for the column points.
// C/D layout: VGPR r, lanes 0-15 -> (M=r, N=lane); lanes 16-31 -> (M=r+8).
// ---------------------------------------------------------------------------
__global__ __launch_bounds__(128) void pool_count_kernel(
    const float* __restrict__ xyz, int* __restrict__ counts) {
  const int lane    = threadIdx.x & 31;
  const int wave    = threadIdx.x >> 5;
  const int b       = blockIdx.y;
  const int i0      = (blockIdx.x * 4 + wave) * 16;
  const int lidx    = lane & 15;
  const int halfSel = lane >> 4;

  const float* X = xyz + (size_t)b * 3 * N_PTS;

  // A-matrix fragment for this row tile
  float ax = X[0 * N_PTS + i0 + lidx];
  float ay = X[1 * N_PTS + i0 + lidx];
  float az = X[2 * N_PTS + i0 + lidx];
  v2f A;
  A.x = halfSel ? az : ax;
  A.y = halfSel ? 0.0f : ay;

  // |xi|^2 for the 8 rows this lane's accumulator VGPRs cover
  float visq[8];
#pragma unroll
  for (int r = 0; r < 8; ++r) {
    int row = i0 + r + halfSel * 8;
    float x = X[0 * N_PTS + row];
    float y = X[1 * N_PTS + row];
    float z = X[2 * N_PTS + row];
    visq[r] = x * x + y * y + z * z;
  }

  int cnt[8] = {0, 0, 0, 0, 0, 0, 0, 0};

  for (int j0 = 0; j0 < N_PTS; j0 += 16) {
    float bx = X[0 * N_PTS + j0 + lidx];
    float by = X[1 * N_PTS + j0 + lidx];
    float bz = X[2 * N_PTS + j0 + lidx];
    if (j0 + 16 < N_PTS) {  // hint next tile into cache (global_prefetch_b8)
      __builtin_prefetch(&X[0 * N_PTS + j0 + 16 + lidx], 0, 0);
    }
    v2f Bm;
    Bm.x = halfSel ? bz : bx;
    Bm.y = halfSel ? 0.0f : by;

    v8f C = {};
    // D = A x B : 16x16 tile of dot products xi . xj
    C = __builtin_amdgcn_wmma_f32_16x16x4_f32(
        /*neg_a=*/false, A, /*neg_b=*/false, Bm,
        /*c_mod=*/(short)0, C, /*reuse_a=*/false, /*reuse_b=*/false);

    float vjsq = bx * bx + by * by + bz * bz;  // per-lane |xj|^2 (N = lidx)
#pragma unroll
    for (int r = 0; r < 8; ++r) {
      float d2 = visq[r] + vjsq - 2.0f * C[r];
      cnt[r] += (d2 < R2 && d2 > RMIN2) ? 1 : 0;
    }
  }

  // Reduce counts over the N dimension: sum across each 16-lane half
#pragma unroll
  for (int r = 0; r < 8; ++r) {
#pragma unroll
    for (int m = 1; m < 16; m <<= 1) cnt[r] += __shfl_xor(cnt[r], m, 32);
  }

  if (lidx == 0) {
#pragma unroll
    for (int r = 0; r < 8; ++r)
      counts[b * N_PTS + i0 + r + halfSel * 8] = cnt[r];
  }
}

// ---------------------------------------------------------------------------
// Top-k by counting sort (deterministic, matches top_k's index-stable order)
// ---------------------------------------------------------------------------
__global__ void pool_zero_hist_kernel(int* __restrict__ hist) {
  int i = blockIdx.x * blockDim.x + threadIdx.x;
  if (i < BATCH * NBINS) hist[i] = 0;
}

__global__ void pool_hist_kernel(const int* __restrict__ counts,
                                 int* __restrict__ hist) {
  int i = blockIdx.x * blockDim.x + threadIdx.x;
  if (i < BATCH * N_PTS) {
    int b = i / N_PTS;
    atomicAdd(&hist[b * NBINS + counts[i]], 1);
  }
}

__global__ void pool_suffix_kernel(const int* __restrict__ hist,
                                   int* __restrict__ startOff) {
  int b = blockIdx.x;
  if (threadIdx.x == 0) {
    int acc = 0;
    for (int c = NBINS - 1; c >= 0; --c) {
      startOff[b * NBINS + c] = acc;
      acc += hist[b * NBINS + c];
    }
  }
}

// One wave per (batch, count-bin): compact matching indices in index order
// using wave32 ballots; write only final positions < KOUT.
__global__ __launch_bounds__(32) void pool_select_kernel(
    const int* __restrict__ counts, const int* __restrict__ startOff,
    int* __restrict__ topIdx) {
  int c = blockIdx.x;
  int b = blockIdx.y;
  int base = startOff[b * NBINS + c];
  if (base >= KOUT) return;
  int lane = threadIdx.x;
  const int* cb = counts + b * N_PTS;
  for (int p = 0; p < N_PTS; p += 32) {
    bool match = (cb[p + lane] == c);
    unsigned mask = (unsigned)__ballot(match);
    if (match) {
      int pos = base + __popc(mask & ((1u << lane) - 1u));
      if (pos < KOUT) topIdx[b * KOUT + pos] = p + lane;
    }
    base += __popc(mask);
    if (base >= KOUT) break;  // wave-uniform
  }
}

// ---------------------------------------------------------------------------
// Gather: out = [new_xyz (B,K,3)] ++ [new_points (B,K,64)]
// ---------------------------------------------------------------------------
__global__ void pool_gather_kernel(const float* __restrict__ xyz,
                                   const float* __restrict__ points,
                                   const int* __restrict__ topIdx,
                                   float* __restrict__ out) {
  int t = blockIdx.x * blockDim.x + threadIdx.x;
  if (t >= BATCH * KOUT) return;
  int b = t / KOUT;
  int k = t % KOUT;
  int idx = topIdx[b * KOUT + k];

  const float* X = xyz + (size_t)b * 3 * N_PTS;
  float* oxyz = out + ((size_t)b * KOUT + k) * 3;
#pragma unroll
  for (int c = 0; c < 3; ++c) oxyz[c] = X[(size_t)c * N_PTS + idx];

  const float* P = points + (size_t)b * FEAT * N_PTS;
  float* op = out + (size_t)BATCH * KOUT * 3 + ((size_t)b * KOUT + k) * FEAT;
#pragma unroll 4
  for (int d = 0; d < FEAT; ++d) op[d] = P[(size_t)d * N_PTS + idx];
}

extern "C" void kernel_launch(void* const* d_in, const int* in_sizes, int n_in,
                              void* d_out, int out_size, void* d_ws,
                              size_t ws_size, hipStream_t stream) {
  const float* xyz    = (const float*)d_in[0];  // [4,3,4096,1] f32
  const float* points = (const float*)d_in[1];  // [4,64,4096,1] f32
  float* out = (float*)d_out;

  int* counts   = (int*)d_ws;                 // B*N
  int* hist     = counts + BATCH * N_PTS;     // B*NBINS
  int* startOff = hist + BATCH * NBINS;       // B*NBINS
  int* topIdx   = startOff + BATCH * NBINS;   // B*KOUT

  pool_count_kernel<<<dim3(N_PTS / 64, BATCH), 128, 0, stream>>>(xyz, counts);
  pool_zero_hist_kernel<<<(BATCH * NBINS + 255) / 256, 256, 0, stream>>>(hist);
  pool_hist_kernel<<<(BATCH * N_PTS + 255) / 256, 256, 0, stream>>>(counts,
                                                                    hist);
  pool_suffix_kernel<<<BATCH, 32, 0, stream>>>(hist, startOff);
  pool_select_kernel<<<dim3(NBINS, BATCH), 32, 0, stream>>>(counts, startOff,
                                                            topIdx);
  pool_gather_kernel<<<(BATCH * KOUT + 63) / 64, 64, 0, stream>>>(
      xyz, points, topIdx, out);
}